// SphericalIoULoss_49821620633795
// MI455X (gfx1250) — compile-verified
//
#include <hip/hip_runtime.h>
#include <cstdint>
#include <cstddef>

// ---------------------------------------------------------------------------
// Spherical IoU loss, MI455X (gfx1250).
// Memory-bound: 160 MB read @ 23.3 TB/s => ~7 us floor. To cover ~800ns HBM
// latency at full bandwidth we need ~19 MB in flight device-wide, so:
//   - 512-item tiles (exactly 5 async b128 copies per thread, wave-uniform)
//   - double-buffered pipeline: issue tile t+1, then s_wait_asynccnt 5 for
//     tile t (async loads retire in order) => ~5 KB in flight per wave.
// ---------------------------------------------------------------------------

#define TILE_ITEMS        512
#define TILE_FLOATS       (TILE_ITEMS * 5)        // 2560 floats per array
#define TILE_BYTES        (TILE_FLOATS * 4)       // 10240 bytes (16B aligned)
#define CHUNKS_PER_ARRAY  (TILE_BYTES / 16)       // 640 x b128 chunks
#define NTHREADS          256
#define CHUNKS_PER_THREAD 5                       // (2*640)/256, wave-uniform
#define MAX_BLOCKS        2048

#define AS1 __attribute__((address_space(1)))
#define AS3 __attribute__((address_space(3)))

typedef int v4i __attribute__((vector_size(16)));

// ---- CDNA5 async copy: global -> LDS, 16 bytes per lane --------------------
__device__ __forceinline__ void async_copy_b128(const float* __restrict__ base,
                                                uint32_t byteOff, float* ldsPtr) {
#if defined(__HIP_DEVICE_COMPILE__)
#if __has_builtin(__builtin_amdgcn_global_load_async_to_lds_b128)
  __builtin_amdgcn_global_load_async_to_lds_b128(
      (AS1 v4i*)(uintptr_t)((const char*)base + byteOff),
      (AS3 v4i*)(uint32_t)(uintptr_t)ldsPtr,
      /*imm offset*/ 0, /*cpol*/ 0);
#else
  uint32_t ldsOff = (uint32_t)(uintptr_t)ldsPtr; // low 32 bits of generic = LDS offset
  asm volatile("global_load_async_to_lds_b128 %0, %1, %2"
               :
               : "v"(ldsOff), "v"(byteOff), "s"(base)
               : "memory");
#endif
#else
  (void)base; (void)byteOff; (void)ldsPtr;
#endif
}

template <int N>
__device__ __forceinline__ void wait_async_le() {
#if defined(__HIP_DEVICE_COMPILE__)
#if __has_builtin(__builtin_amdgcn_s_wait_asynccnt)
  __builtin_amdgcn_s_wait_asynccnt(N);
#else
  asm volatile("s_wait_asynccnt %0" :: "i"(N) : "memory");
#endif
#endif
}

// ---- math ------------------------------------------------------------------
// asin(x) for |x| <= ~0.27 (guaranteed by fov ranges): odd series, err < 2e-7.
__device__ __forceinline__ float asin_small(float x) {
  float x2 = x * x;
  float p = fmaf(x2, 0.03038196f, 0.04464286f);   // 105/3456, 15/336
  p = fmaf(x2, p, 0.075f);                        // 3/40
  p = fmaf(x2, p, 0.16666667f);                   // 1/6
  return fmaf(x * x2, p, x);
}

// Reference: 4*acos(clip(-sin(a/2)sin(b/2),-1,1)) - 2*pi == 4*asin(sin(a/2)sin(b/2))
__device__ __forceinline__ float rect_area4(float a, float b) {
  float s = __sinf(0.5f * a) * __sinf(0.5f * b);
  return 4.0f * asin_small(s);
}

__device__ __forceinline__ float loss_item(float t1, float p1, float a1, float b1,
                                           float t2, float p2, float a2, float b2) {
  float area1 = rect_area4(a1, b1);
  float area2 = rect_area4(a2, b2);
  float dfx = (t2 - t1) * __cosf(0.5f * (p1 + p2));
  float dfy = p2 - p1;
  float iw = fminf(0.5f * a1, dfx + 0.5f * a2) - fmaxf(-0.5f * a1, dfx - 0.5f * a2);
  iw = fmaxf(iw, 0.0f);
  float ih = fminf(0.5f * b1, dfy + 0.5f * b2) - fmaxf(-0.5f * b1, dfy - 0.5f * b2);
  ih = fmaxf(ih, 0.0f);
  float inter = rect_area4(iw, ih);
  float uni = area1 + area2 - inter;
  float iou = inter / fmaxf(uni, 1e-8f);
  return 1.0f - iou;
}

// Issue one tile's async copies. 1280 b128 chunks, 5 per thread; the
// preds/targets split (c < 640) falls on a wave boundary (tid<128), so every
// wave issues exactly 5 async instructions -> wait arithmetic is exact.
__device__ __forceinline__ void issue_tile(const float* __restrict__ preds,
                                           const float* __restrict__ targets,
                                           uint32_t tileByte,
                                           float* bufP, float* bufT, int tid) {
#pragma unroll
  for (int k = 0; k < CHUNKS_PER_THREAD; ++k) {
    int c = tid + k * NTHREADS;                      // 0..1279
    if (c < CHUNKS_PER_ARRAY) {
      async_copy_b128(preds, tileByte + (uint32_t)c * 16u, bufP + c * 4);
    } else {
      int c2 = c - CHUNKS_PER_ARRAY;
      async_copy_b128(targets, tileByte + (uint32_t)c2 * 16u, bufT + c2 * 4);
    }
  }
}

// ---- kernel 1: double-buffered streaming + deterministic partial sums ------
__global__ void __launch_bounds__(NTHREADS)
sph_iou_partial_kernel(const float* __restrict__ preds,
                       const float* __restrict__ targets,
                       float* __restrict__ partials,
                       int nItems, int nFullTiles) {
  // [buffer][P=0 / T=1][floats] : 2 * 2 * 10240 B = 40 KB
  __shared__ float lds[2][2][TILE_FLOATS];
  __shared__ float red[NTHREADS];

  const int tid = threadIdx.x;
  float acc = 0.0f;

  int tile = blockIdx.x;
  int cur = 0;
  if (tile < nFullTiles) {
    issue_tile(preds, targets, (uint32_t)tile * TILE_BYTES,
               lds[0][0], lds[0][1], tid);
  }

  while (tile < nFullTiles) {
    const int nextTile = tile + (int)gridDim.x;
    const bool haveNext = nextTile < nFullTiles;

    if (haveNext) {
      issue_tile(preds, targets, (uint32_t)nextTile * TILE_BYTES,
                 lds[cur ^ 1][0], lds[cur ^ 1][1], tid);
      wait_async_le<CHUNKS_PER_THREAD>();   // tile `tile` landed; next 5 fly on
    } else {
      wait_async_le<0>();
    }
    __syncthreads();                        // all waves' copies visible

    const float* bp = lds[cur][0];
    const float* bt = lds[cur][1];
    // stride-5 DWORD access, lanes differ by 5 dwords: conflict-free banks
    {
      const float* p = &bp[tid * 5];
      const float* t = &bt[tid * 5];
      acc += loss_item(p[0], p[1], p[2], p[3], t[0], t[1], t[2], t[3]);
    }
    {
      const float* p = &bp[(tid + NTHREADS) * 5];
      const float* t = &bt[(tid + NTHREADS) * 5];
      acc += loss_item(p[0], p[1], p[2], p[3], t[0], t[1], t[2], t[3]);
    }
    __syncthreads();                        // done reading before buffer reuse

    tile = nextTile;
    cur ^= 1;
  }

  // tail items (nItems not multiple of TILE_ITEMS) -> block 0, plain loads
  if (blockIdx.x == 0) {
    for (int i = nFullTiles * TILE_ITEMS + tid; i < nItems; i += NTHREADS) {
      const float* p = &preds[(size_t)i * 5];
      const float* t = &targets[(size_t)i * 5];
      acc += loss_item(p[0], p[1], p[2], p[3], t[0], t[1], t[2], t[3]);
    }
  }

  // deterministic block tree-reduce
  red[tid] = acc;
  __syncthreads();
  for (int s = NTHREADS / 2; s > 0; s >>= 1) {
    if (tid < s) red[tid] += red[tid + s];
    __syncthreads();
  }
  if (tid == 0) partials[blockIdx.x] = red[0];
}

// ---- kernel 2: deterministic final reduce + mean ---------------------------
__global__ void __launch_bounds__(NTHREADS)
sph_iou_finalize_kernel(const float* __restrict__ partials, int nPartials,
                        float* __restrict__ out, float invN) {
  __shared__ float red[NTHREADS];
  const int tid = threadIdx.x;
  float acc = 0.0f;
  for (int i = tid; i < nPartials; i += NTHREADS) acc += partials[i];
  red[tid] = acc;
  __syncthreads();
  for (int s = NTHREADS / 2; s > 0; s >>= 1) {
    if (tid < s) red[tid] += red[tid + s];
    __syncthreads();
  }
  if (tid == 0) out[0] = red[0] * invN;
}

// ---------------------------------------------------------------------------
extern "C" void kernel_launch(void* const* d_in, const int* in_sizes, int n_in,
                              void* d_out, int out_size, void* d_ws, size_t ws_size,
                              hipStream_t stream) {
  const float* preds   = (const float*)d_in[0];
  const float* targets = (const float*)d_in[1];
  float* out = (float*)d_out;
  float* partials = (float*)d_ws;

  const int nItems = in_sizes[0] / 5;
  const int nFullTiles = nItems / TILE_ITEMS;

  int grid = nFullTiles < MAX_BLOCKS ? nFullTiles : MAX_BLOCKS;
  if (grid < 1) grid = 1;

  sph_iou_partial_kernel<<<grid, NTHREADS, 0, stream>>>(
      preds, targets, partials, nItems, nFullTiles);

  const float invN = (nItems > 0) ? (float)(1.0 / (double)nItems) : 0.0f;
  sph_iou_finalize_kernel<<<1, NTHREADS, 0, stream>>>(partials, grid, out, invN);
}